// SparseConvBlock_22617297781334
// MI455X (gfx1250) — compile-verified
//
#include <hip/hip_runtime.h>

// SparseConvBlock: 81-tap gather-GEMM (f32 WMMA) + fused BatchNorm stats + BN/ReLU apply.
// Tile: 256 points x 64 channels per 256-thread block; each wave32 owns 32x64
// (2 M-tiles x 4 N-tiles = 8 accumulators) -> 0.75 ds_load_b64 per v_wmma.

#define PTS_TILE 256
#define CIN      64
#define COUT     64
#define KVOL     81
#define LDS_STR  68      // 64 + 4 pad -> conflict-free strided ds_load_b64 fragments
#define BN_EPS   1e-5f

typedef float v2f __attribute__((ext_vector_type(2)));
typedef float v8f __attribute__((ext_vector_type(8)));

__global__ __launch_bounds__(256) void zero_sums_kernel(float* gsums) {
  int t = threadIdx.x;
  if (t < 128) gsums[t] = 0.f;
}

__global__ __launch_bounds__(256) void spconv_wmma_kernel(
    const float* __restrict__ feat,      // [N, 64]
    const float* __restrict__ W,         // [81, 64, 64]
    const int*   __restrict__ nbr,       // [81, N]
    float*       __restrict__ conv_out,  // [N, 64] (staged in d_out)
    float*       __restrict__ gsums,     // [128]: sum[64], sumsq[64]
    int nPts) {
  __shared__ float sfeat[PTS_TILE * LDS_STR];   // gathered tile, row-padded
  __shared__ float swt[COUT * LDS_STR];         // W[k] transposed: swt[n][c]
  __shared__ float ssum[128];

  const int tid    = threadIdx.x;
  const int wave   = tid >> 5;
  const int lane   = tid & 31;
  const int lane16 = lane & 15;
  const int koff   = (lane >> 4) << 1;  // 0 lanes 0-15, 2 lanes 16-31 (A/B VGPR layout)
  const int m0     = wave << 5;         // this wave's 32-point strip
  const int base   = blockIdx.x * PTS_TILE;

  if (tid < 128) ssum[tid] = 0.f;

  v8f acc[2][4];                        // [M-tile][N-tile]
  #pragma unroll
  for (int mt = 0; mt < 2; ++mt)
    #pragma unroll
    for (int t = 0; t < 4; ++t)
      #pragma unroll
      for (int r = 0; r < 8; ++r) acc[mt][t][r] = 0.f;

  for (int k = 0; k < KVOL; ++k) {
    __syncthreads();  // protect LDS reuse from previous iteration's compute

    // ---- gather 256 feature rows: 1 thread per point, 64 floats each ----
    {
      const int gp = base + tid;
      float4* dst = (float4*)&sfeat[tid * LDS_STR];
      int idx = nPts;  // pad
      if (gp < nPts) {
        idx = nbr[(size_t)k * nPts + gp];
        if (k + 1 < KVOL)  // stream next tap's index row into cache
          __builtin_prefetch(&nbr[(size_t)(k + 1) * nPts + gp], 0, 1);
      }
      if (idx >= 0 && idx < nPts) {
        const float4* src = (const float4*)(feat + (size_t)idx * CIN);
        #pragma unroll
        for (int j = 0; j < 16; ++j) dst[j] = src[j];
      } else {
        #pragma unroll
        for (int j = 0; j < 16; ++j) dst[j] = make_float4(0.f, 0.f, 0.f, 0.f);
      }
    }

    // ---- load W[k] transposed into LDS: swt[n*LDS_STR + c] = W[k][c][n] ----
    {
      const float* Wk = W + (size_t)k * (CIN * COUT);
      #pragma unroll
      for (int i = tid * 4; i < CIN * COUT; i += 256 * 4) {
        float4 w = *(const float4*)(Wk + i);
        int c = i >> 6, n = i & 63;
        swt[(n + 0) * LDS_STR + c] = w.x;
        swt[(n + 1) * LDS_STR + c] = w.y;
        swt[(n + 2) * LDS_STR + c] = w.z;
        swt[(n + 3) * LDS_STR + c] = w.w;
      }
    }
    __syncthreads();

    // ---- WMMA: [32 x 64] feature strip  x  [64 x 64] weights ----
    const float* arow0 = &sfeat[(m0 + lane16) * LDS_STR];        // M-tile 0
    const float* arow1 = arow0 + 16 * LDS_STR;                   // M-tile 1
    const float* brow  = &swt[lane16 * LDS_STR];
    #pragma unroll
    for (int kk = 0; kk < CIN; kk += 4) {
      v2f a0 = *(const v2f*)(arow0 + kk + koff);
      v2f a1 = *(const v2f*)(arow1 + kk + koff);
      #pragma unroll
      for (int t = 0; t < 4; ++t) {
        v2f b = *(const v2f*)(brow + (t << 4) * LDS_STR + kk + koff);
        acc[0][t] = __builtin_amdgcn_wmma_f32_16x16x4_f32(
            false, a0, false, b, (short)0, acc[0][t], false, false);
        acc[1][t] = __builtin_amdgcn_wmma_f32_16x16x4_f32(
            false, a1, false, b, (short)0, acc[1][t], false, false);
      }
    }
  }

  // ---- write conv tile + fused BN partial sums ----
  __syncthreads();
  const int mhi = (lane >> 4) << 3;  // C/D layout: lanes 16-31 hold M = r + 8
  #pragma unroll
  for (int t = 0; t < 4; ++t) {
    const int ch = (t << 4) + lane16;
    float s = 0.f, s2 = 0.f;
    #pragma unroll
    for (int mt = 0; mt < 2; ++mt) {
      #pragma unroll
      for (int r = 0; r < 8; ++r) {
        float v = acc[mt][t][r];
        s += v; s2 += v * v;           // padded/tail rows are exactly 0 -> no bias
        int gp = base + m0 + (mt << 4) + mhi + r;
        if (gp < nPts) conv_out[(size_t)gp * COUT + ch] = v;
      }
    }
    atomicAdd(&ssum[ch], s);
    atomicAdd(&ssum[64 + ch], s2);
  }
  __syncthreads();
  if (tid < 128) atomicAdd(&gsums[tid], ssum[tid]);
}

__global__ __launch_bounds__(256) void bn_relu_kernel(
    float* __restrict__ out, const float* __restrict__ gsums,
    const float* __restrict__ gamma, const float* __restrict__ beta, int nPts) {
  size_t i = ((size_t)blockIdx.x * blockDim.x + threadIdx.x) * 4;
  size_t total = (size_t)nPts * COUT;
  if (i >= total) return;
  const int c0 = (int)(i & 63);
  const float invN = 1.f / (float)nPts;
  float4 x = *(float4*)(out + i);
  float y[4] = {x.x, x.y, x.z, x.w};
  #pragma unroll
  for (int j = 0; j < 4; ++j) {
    int c = c0 + j;
    float mean = gsums[c] * invN;
    float var  = gsums[64 + c] * invN - mean * mean;
    float v = (y[j] - mean) * rsqrtf(var + BN_EPS) * gamma[c] + beta[c];
    y[j] = fmaxf(v, 0.f);
  }
  *(float4*)(out + i) = make_float4(y[0], y[1], y[2], y[3]);
}

extern "C" void kernel_launch(void* const* d_in, const int* in_sizes, int n_in,
                              void* d_out, int out_size, void* d_ws, size_t ws_size,
                              hipStream_t stream) {
  const float* feat  = (const float*)d_in[0];   // features [N,64]
  const float* W     = (const float*)d_in[1];   // W [81,64,64]
  const float* gamma = (const float*)d_in[2];   // [64]
  const float* beta  = (const float*)d_in[3];   // [64]
  const int*   nbr   = (const int*)d_in[4];     // neighbor_idx [81,N]
  float* out   = (float*)d_out;
  float* gsums = (float*)d_ws;                  // 128 floats of scratch
  const int nPts = in_sizes[0] / CIN;

  zero_sums_kernel<<<1, 256, 0, stream>>>(gsums);

  const int blocks = (nPts + PTS_TILE - 1) / PTS_TILE;
  spconv_wmma_kernel<<<blocks, 256, 0, stream>>>(feat, W, nbr, out, gsums, nPts);

  const size_t total4 = ((size_t)nPts * COUT) / 4;
  bn_relu_kernel<<<(int)((total4 + 255) / 256), 256, 0, stream>>>(
      out, gsums, gamma, beta, nPts);
}